// Pair_83811991814342
// MI455X (gfx1250) — compile-verified
//
#include <hip/hip_runtime.h>

// Pairwise feature-concat + IoU ratios (store-bandwidth bound, 547 MB out).
//   in : (1, 1024, 260) fp32, rows [0,512) used; cols 256..259 = bbox x1,y1,x2,y2
//   out: (1, 512*512, 522) fp32, row(p=i*512+j) = [x_i | x_j | r1 | r2]
//
// One wave32 handles TWO consecutive output rows (same i, adjacent j):
//   even row (16B-aligned):  130 aligned b128 stores + b64 ratio tail
//   odd  row (8 mod 16):     b64 head + 130 aligned b128 stores (slots shifted by 2 floats)
// All stores non-temporal (output is write-once streaming; keep input hot in L2).

typedef float v2f __attribute__((ext_vector_type(2)));
typedef float v4f __attribute__((ext_vector_type(4)));

#define PN   512
#define PD   260
#define ROWF 522   // floats per output row

__device__ __forceinline__ void iou_ratios(const v4f bi, const v4f bj,
                                           float ai, float aj,
                                           float& r1, float& r2) {
    const float w = fmaxf(0.0f, fminf(bi.z, bj.z) - fmaxf(bi.x, bj.x));
    const float h = fmaxf(0.0f, fminf(bi.w, bj.w) - fmaxf(bi.y, bj.y));
    const float inter = w * h;
    r1 = inter / ai;   // inter / area_i  (IEEE div, matches jax fp32)
    r2 = inter / aj;   // inter / area_j
}

__global__ __launch_bounds__(256) void pair_iou_concat2_kernel(
    const float* __restrict__ x, float* __restrict__ out)
{
    const unsigned lane = threadIdx.x & 31u;
    const unsigned w    = (blockIdx.x << 3) | (threadIdx.x >> 5); // wave id
    const unsigned p0   = w << 1;          // even pair row; p0,p0+1 share i
    const unsigned i    = p0 >> 9;
    const unsigned j0   = p0 & 511u;

    const float* __restrict__ xi  = x + i  * PD;
    const float* __restrict__ xj0 = x + j0 * PD;
    const float* __restrict__ xj1 = xj0 + PD;   // j1 = j0+1 (never wraps: j0 even)

    // gfx1250 prefetch (global_prefetch_b8)
    __builtin_prefetch(xi, 0, 3);
    __builtin_prefetch(xj0, 0, 3);

    // bbox: one aligned float4 per row (row base + 1024B, 16B aligned)
    const v4f bi  = *(const v4f*)(xi  + PD - 4);
    const v4f bj0 = *(const v4f*)(xj0 + PD - 4);
    const v4f bj1 = *(const v4f*)(xj1 + PD - 4);
    const float ai  = (bi.z  - bi.x )*(bi.w  - bi.y );
    const float aj0 = (bj0.z - bj0.x)*(bj0.w - bj0.y);
    const float aj1 = (bj1.z - bj1.x)*(bj1.w - bj1.y);

    float r1e, r2e, r1o, r2o;
    iou_ratios(bi, bj0, ai, aj0, r1e, r2e);
    iou_ratios(bi, bj1, ai, aj1, r1o, r2o);

    float* __restrict__ oe = out + (size_t)p0 * ROWF; // even row: 16B aligned
    float* __restrict__ oo = oe + ROWF;               // odd row: 8 mod 16

    // ---- even row: floats [0,520) as 130 aligned float4 slots ----
    // slot k4: floats 4k4..4k4+3; k4<65 -> xi (16B aligned), else xj0 (16B aligned)
    #pragma unroll
    for (unsigned it = 0; it < 5; ++it) {
        const unsigned k4 = (it << 5) + lane;
        if (k4 < 130u) {
            const v4f v = (k4 < 65u) ? *(const v4f*)(xi  + 4u * k4)
                                     : *(const v4f*)(xj0 + 4u * k4 - 260u);
            __builtin_nontemporal_store(v, (v4f*)oe + k4);
        }
    }
    if (lane == 0u) {
        v2f t; t.x = r1e; t.y = r2e;                 // even row ratio tail
        __builtin_nontemporal_store(t, (v2f*)(oe + 520));
        const v2f hd = *(const v2f*)xi;              // odd row head: xi[0..1]
        __builtin_nontemporal_store(hd, (v2f*)oo);
    }

    // ---- odd row: floats [2,522) as 130 aligned float4 slots at oo+2 ----
    // slot m: floats 2+4m .. 5+4m, built from two 8B-aligned v2f loads
    v4f* __restrict__ ob = (v4f*)(oo + 2);           // 16B aligned
    #pragma unroll
    for (unsigned it = 0; it < 5; ++it) {
        const unsigned m = (it << 5) + lane;
        if (m < 130u) {
            v2f lo, hi;
            if (m < 64u) {                            // all from xi
                lo = *(const v2f*)(xi + 2u + 4u * m);
                hi = *(const v2f*)(xi + 4u + 4u * m);
            } else if (m == 64u) {                    // xi tail | xj1 head
                lo = *(const v2f*)(xi + 258);
                hi = *(const v2f*)(xj1);
            } else if (m < 129u) {                    // all from xj1
                lo = *(const v2f*)(xj1 + 4u * m - 258u);
                hi = *(const v2f*)(xj1 + 4u * m - 256u);
            } else {                                  // xj1 tail | ratios
                lo = *(const v2f*)(xj1 + 258);
                hi.x = r1o; hi.y = r2o;
            }
            v4f v; v.x = lo.x; v.y = lo.y; v.z = hi.x; v.w = hi.y;
            __builtin_nontemporal_store(v, ob + m);
        }
    }
}

extern "C" void kernel_launch(void* const* d_in, const int* in_sizes, int n_in,
                              void* d_out, int out_size, void* d_ws, size_t ws_size,
                              hipStream_t stream) {
    const float* x = (const float*)d_in[0];   // (1, 1024, 260) fp32
    float* out = (float*)d_out;               // (1, 512*512, 522) fp32

    const unsigned n_waves = (PN * PN) / 2;   // 131072 waves, 2 rows each
    const unsigned blocks  = n_waves / 8;     // 8 waves (256 thr) per block
    pair_iou_concat2_kernel<<<dim3(blocks), dim3(256), 0, stream>>>(x, out);
}